// DotProductAttention_58368605552662
// MI455X (gfx1250) — compile-verified
//
#include <hip/hip_runtime.h>

// fp32 flash attention for MI455X (gfx1250), wave32 + V_WMMA_F32_16X16X4_F32.
// B=16, Q=K=2048, D=128. Working set (50MB) is L2-resident (192MB L2), so
// K/V are streamed with plain global loads; compute bound -> WMMA pipe.

typedef __attribute__((ext_vector_type(2))) float v2f;
typedef __attribute__((ext_vector_type(8))) float v8f;

#define ATTN_NEG -1000000.0f

__global__ __launch_bounds__(128)
void DotProductAttention_58368605552662_kernel(
    const float* __restrict__ Qm, const float* __restrict__ Km,
    const float* __restrict__ Vm, const int* __restrict__ vlens,
    float* __restrict__ Om)
{
    constexpr int Qn = 2048, Kn = 2048, Dn = 128;
    const int lane = threadIdx.x & 31;
    const int wave = threadIdx.x >> 5;
    const int half = lane >> 4;      // 0: lanes 0-15, 1: lanes 16-31
    const int l16  = lane & 15;

    const int b      = blockIdx.x >> 5;   // 32 query-chunks per batch
    const int qchunk = blockIdx.x & 31;
    const int q0     = qchunk * 64 + wave * 16;   // this wave's 16-query tile
    const int vlen   = vlens[b];

    const float* Qb = Qm + (size_t)b * Qn * Dn;
    const float* Kb = Km + (size_t)b * Kn * Dn;
    const float* Vb = Vm + (size_t)b * Kn * Dn;
    float*       Ob = Om + (size_t)b * Qn * Dn;

    // ---- Q operand, resident for whole K loop ----
    // B-matrix (4x16, K=d-chunk, N=query) of S^T = Ktile x Q^T:
    // lane holds Q[q0 + l16][4c + 2*half + {0,1}]  (v0=K0/K2, v1=K1/K3)
    v2f qreg[32];
    const float* qrow = Qb + (size_t)(q0 + l16) * Dn + 2 * half;
#pragma unroll
    for (int c = 0; c < 32; ++c)
        qreg[c] = *(const v2f*)(qrow + 4 * c);

    // ---- accumulators ----
    v8f oacc[8];                       // O tile: 8 d-subtiles of 16 cols
#pragma unroll
    for (int t = 0; t < 8; ++t) oacc[t] = (v8f)0.0f;
    float m_run = -3.0e38f;            // running row max
    float l_run = 0.0f;                // running row sum

    const float scale = 0.08838834764831845f;   // 1/sqrt(128)

    for (int kt = 0; kt < Kn / 16; ++kt) {
        const int krow0 = kt * 16;

        // prefetch next K tile rows (uniform branch; lands in L2/L0)
        if (kt + 1 < Kn / 16)
            __builtin_prefetch(Kb + (size_t)(krow0 + 16 + l16) * Dn, 0, 0);

        // ---- GEMM 1: S^T (16 keys x 16 queries) = Ktile(16x128) x Q^T ----
        // A-matrix: lane holds K[krow0 + l16][4c + 2*half + {0,1}]
        v8f sacc = (v8f)0.0f;
        const float* krow = Kb + (size_t)(krow0 + l16) * Dn + 2 * half;
#pragma unroll
        for (int c = 0; c < 32; ++c) {
            v2f a = *(const v2f*)(krow + 4 * c);
            sacc = __builtin_amdgcn_wmma_f32_16x16x4_f32(
                false, a, false, qreg[c], (short)0, sacc, false, false);
        }

        // ---- scale + key mask + online softmax ----
        // S^T C-layout: VGPR r holds key = krow0 + r + 8*half, query = l16
        float p[8];
        float tmax = -3.0e38f;
#pragma unroll
        for (int r = 0; r < 8; ++r) {
            float s = sacc[r] * scale;
            int kg = krow0 + r + 8 * half;
            if (kg >= vlen) s = ATTN_NEG;   // finite mask, matches reference
            p[r] = s;
            tmax = fmaxf(tmax, s);
        }
        tmax = fmaxf(tmax, __shfl_xor(tmax, 16, 32));   // combine lane halves
        const float m_new = fmaxf(m_run, tmax);
        const float alpha = __expf(m_run - m_new);
        float tsum = 0.0f;
#pragma unroll
        for (int r = 0; r < 8; ++r) {
            p[r] = __expf(p[r] - m_new);
            tsum += p[r];
        }
        tsum += __shfl_xor(tsum, 16, 32);
        l_run = l_run * alpha + tsum;
        m_run = m_new;

        // rescale O accumulators: O row = query = r + 8*half, alpha lives at
        // lanes whose l16 == query -> broadcast from lane (r + 8*half)
#pragma unroll
        for (int r = 0; r < 8; ++r) {
            float ab = __shfl(alpha, r + 8 * half, 32);
#pragma unroll
            for (int t = 0; t < 8; ++t)
                oacc[t][r] = oacc[t][r] * ab;
        }

        // ---- convert P (S^T C-layout) to A-layout for P x V ----
        // lane half 0 needs keys {4j,4j+1}; half 1 needs keys {4j+2,4j+3}
        float pq[8];
#pragma unroll
        for (int r = 0; r < 8; ++r)
            pq[r] = __shfl_xor(p[r], 16, 32);
        v2f aP[4];
        aP[0] = v2f{ half ? pq[2] : p[0],  half ? pq[3] : p[1] };
        aP[1] = v2f{ half ? pq[6] : p[4],  half ? pq[7] : p[5] };
        aP[2] = v2f{ half ? p[2]  : pq[0], half ? p[3]  : pq[1] };
        aP[3] = v2f{ half ? p[6]  : pq[4], half ? p[7]  : pq[5] };

        // ---- GEMM 2: O(16q x 128d) += P(16q x 16k) x V(16k x 128d) ----
        // B-matrix per (j,t): lane holds V[krow0+4j+2*half + {0,1}][t*16 + l16]
#pragma unroll
        for (int j = 0; j < 4; ++j) {
            const float* vr = Vb + (size_t)(krow0 + 4 * j + 2 * half) * Dn + l16;
#pragma unroll
            for (int t = 0; t < 8; ++t) {
                v2f bv = v2f{ vr[t * 16], vr[Dn + t * 16] };
                oacc[t] = __builtin_amdgcn_wmma_f32_16x16x4_f32(
                    false, aP[j], false, bv, (short)0, oacc[t], false, false);
            }
        }
    }

    // ---- normalize by l and store ----
    const float linv = 1.0f / l_run;
#pragma unroll
    for (int r = 0; r < 8; ++r) {
        const float lb = __shfl(linv, r + 8 * half, 32);
        const int qrow_o = q0 + r + 8 * half;
        float* orow = Ob + (size_t)qrow_o * Dn + l16;
#pragma unroll
        for (int t = 0; t < 8; ++t)
            orow[t * 16] = oacc[t][r] * lb;
    }
}

extern "C" void kernel_launch(void* const* d_in, const int* in_sizes, int n_in,
                              void* d_out, int out_size, void* d_ws, size_t ws_size,
                              hipStream_t stream) {
    const float* q  = (const float*)d_in[0];
    const float* k  = (const float*)d_in[1];
    const float* v  = (const float*)d_in[2];
    const int*   vl = (const int*)d_in[3];
    float* out = (float*)d_out;

    // B * (Q/64) blocks, 4 wave32s each (one 16-query tile per wave)
    dim3 grid(16 * (2048 / 64));
    dim3 block(128);
    DotProductAttention_58368605552662_kernel<<<grid, block, 0, stream>>>(q, k, v, vl, out);
}